// ST_GAT_85650237817560
// MI455X (gfx1250) — compile-verified
//
#include <hip/hip_runtime.h>
#include <math.h>

#define NN 50000
#define NE 1600000
#define H1 32
#define H2 128

typedef __attribute__((ext_vector_type(2))) float v2f;
typedef __attribute__((ext_vector_type(8))) float v8f;

#if __has_builtin(__builtin_amdgcn_global_load_async_to_lds_b32) && __has_builtin(__builtin_amdgcn_s_wait_asynccnt)
#define ASYNC_XT 1
typedef __attribute__((address_space(1))) int* gptr_i;
typedef __attribute__((address_space(3))) int* lptr_i;
#else
#define ASYNC_XT 0
#endif

__device__ __forceinline__ float sigmoidf_(float x){ return 1.0f/(1.0f+expf(-x)); }
// order-preserving float<->uint mapping for atomicMax-based segment max
__device__ __forceinline__ unsigned enc_f(float f){ unsigned u=__float_as_uint(f); return (u&0x80000000u)?~u:(u|0x80000000u); }
__device__ __forceinline__ float dec_f(unsigned u){ return (u&0x80000000u)?__uint_as_float(u^0x80000000u):__uint_as_float(~u); }

// ---------------- GEMM: C[M,Nc] = A[M,K] @ B[K,Nc],  B given row-major [K,Nc] (for x@W) ----------
__global__ void k_gemm_xw(const float* __restrict__ A, const float* __restrict__ B,
                          float* __restrict__ C)
{
    const int K = 12, Nc = 64, tiles_n = 4, tiles = (NN/16)*tiles_n;
    int gw   = (blockIdx.x * blockDim.x + threadIdx.x) >> 5;
    int lane = threadIdx.x & 31;
    if (gw >= tiles) return;                       // whole-wave uniform exit: EXEC stays all-ones
    int tm = gw / tiles_n, tn = gw % tiles_n;
    int half = lane >> 4, l16 = lane & 15;
    int row = tm*16 + l16, col = tn*16 + l16;
    v8f acc = {};
    #pragma unroll
    for (int k0 = 0; k0 < K; k0 += 4) {
        int ka = k0 + 2*half;
        v2f a = *(const v2f*)(A + row*K + ka);     // A 16x4 layout: VGPR0/1 = K=ka,ka+1
        v2f b; b.x = B[ka*Nc + col]; b.y = B[(ka+1)*Nc + col];
        acc = __builtin_amdgcn_wmma_f32_16x16x4_f32(false, a, false, b, (short)0, acc, false, false);
    }
    #pragma unroll
    for (int v = 0; v < 8; v++) {
        int r = tm*16 + v + 8*half;
        C[r*Nc + col] = acc[v];
    }
}

// ---------------- GEMM: C[M,Nc] = A[M,K] @ Bt[Nc,K]^T  (+ optional column bias) -----------------
// K is a template parameter: full unroll -> long s_clause load batches feeding the WMMA chain.
template<int K>
__global__ void k_gemm_ABt(const float* __restrict__ A, const float* __restrict__ Bt,
                           const float* __restrict__ bias0, const float* __restrict__ bias1,
                           float* __restrict__ C, int M, int Nc)
{
    int tiles_n = Nc >> 4;
    int tiles   = (M >> 4) * tiles_n;
    int gw   = (blockIdx.x * blockDim.x + threadIdx.x) >> 5;
    int lane = threadIdx.x & 31;
    if (gw >= tiles) return;
    int tm = gw / tiles_n, tn = gw % tiles_n;
    int half = lane >> 4, l16 = lane & 15;
    int row = tm*16 + l16, col = tn*16 + l16;
    v8f acc = {};
    #pragma unroll
    for (int k0 = 0; k0 < K; k0 += 4) {
        int ka = k0 + 2*half;
        v2f a = *(const v2f*)(A  + row*K + ka);
        v2f b = *(const v2f*)(Bt + col*K + ka);    // Bt contiguous in k
        acc = __builtin_amdgcn_wmma_f32_16x16x4_f32(false, a, false, b, (short)0, acc, false, false);
    }
    float bs = 0.0f;
    if (bias0) bs += bias0[col];
    if (bias1) bs += bias1[col];
    #pragma unroll
    for (int v = 0; v < 8; v++) {
        int r = tm*16 + v + 8*half;
        C[r*Nc + col] = acc[v] + bs;
    }
}

// ---------------- per-(node,head) attention logits ----------------
__global__ void k_alpha(const float* __restrict__ h, const float* __restrict__ a_src,
                        const float* __restrict__ a_dst, float* __restrict__ as_,
                        float* __restrict__ ad_)
{
    int idx = blockIdx.x * blockDim.x + threadIdx.x;
    if (idx >= NN*8) return;
    int n = idx >> 3, hd = idx & 7;
    const float* hp = h + n*64 + hd*8;
    float s = 0.f, d = 0.f;
    #pragma unroll
    for (int c = 0; c < 8; c++) { float v = hp[c]; s += v*a_src[hd*8+c]; d += v*a_dst[hd*8+c]; }
    as_[idx] = s; ad_[idx] = d;
}

// ---------------- edge pass 1: segment max (encoded u32 atomicMax; buffers L2-resident) ---------
__global__ void k_edge_max(const int* __restrict__ ei, const float* __restrict__ as_,
                           const float* __restrict__ ad_, unsigned* __restrict__ menc)
{
    int e = blockIdx.x * blockDim.x + threadIdx.x;
    if (e >= NE) return;
    int s = ei[e], d = ei[NE + e];
    #pragma unroll
    for (int hd = 0; hd < 8; hd++) {
        float v = as_[s*8+hd] + ad_[d*8+hd];
        v = (v > 0.f) ? v : 0.2f*v;
        atomicMax(&menc[d*8+hd], enc_f(v));
    }
}

// ---------------- edge pass 2: segment sum of exp(e - m) ----------------
__global__ void k_edge_sum(const int* __restrict__ ei, const float* __restrict__ as_,
                           const float* __restrict__ ad_, const unsigned* __restrict__ menc,
                           float* __restrict__ denom)
{
    int e = blockIdx.x * blockDim.x + threadIdx.x;
    if (e >= NE) return;
    int s = ei[e], d = ei[NE + e];
    #pragma unroll
    for (int hd = 0; hd < 8; hd++) {
        float v = as_[s*8+hd] + ad_[d*8+hd];
        v = (v > 0.f) ? v : 0.2f*v;
        atomicAdd(&denom[d*8+hd], expf(v - dec_f(menc[d*8+hd])));
    }
}

// ---------------- edge pass 3: scatter alpha * h[src] into out[dst] ----------------
__global__ void k_edge_msg(const int* __restrict__ ei, const float* __restrict__ as_,
                           const float* __restrict__ ad_, const unsigned* __restrict__ menc,
                           const float* __restrict__ denom, const float* __restrict__ h,
                           float* __restrict__ outb)
{
    long long idx = (long long)blockIdx.x * blockDim.x + threadIdx.x;
    if (idx >= (long long)NE*8) return;
    int e = (int)(idx >> 3), hd = (int)(idx & 7);
    int s = ei[e], d = ei[NE + e];
    float v = as_[s*8+hd] + ad_[d*8+hd];
    v = (v > 0.f) ? v : 0.2f*v;
    float al = expf(v - dec_f(menc[d*8+hd])) / (denom[d*8+hd] + 1e-16f);
    const float* hp = h + s*64 + hd*8;
    float* op = outb + d*64 + hd*8;
    #pragma unroll
    for (int c = 0; c < 8; c++) atomicAdd(op + c, al * hp[c]);
}

// ---------------- +bias, leaky_relu(0.01), log_softmax over 64 cols (one wave per node) --------
__global__ void k_finalize(float* __restrict__ buf, const float* __restrict__ gb)
{
    int w = threadIdx.x >> 5, lane = threadIdx.x & 31;
    int node = blockIdx.x * (blockDim.x >> 5) + w;
    if (node >= NN) return;
    float* r = buf + node*64;
    float v0 = r[lane]      + gb[lane];
    float v1 = r[lane + 32] + gb[lane + 32];
    v0 = (v0 > 0.f) ? v0 : 0.01f*v0;
    v1 = (v1 > 0.f) ? v1 : 0.01f*v1;
    float m = fmaxf(v0, v1);
    #pragma unroll
    for (int o = 16; o; o >>= 1) m = fmaxf(m, __shfl_xor(m, o, 32));
    float ssum = expf(v0 - m) + expf(v1 - m);
    #pragma unroll
    for (int o = 16; o; o >>= 1) ssum += __shfl_xor(ssum, o, 32);
    float lse = logf(ssum);
    r[lane]      = v0 - m - lse;
    r[lane + 32] = v1 - m - lse;
}

// ---------------- LSTM1 recurrence (input proj G1 precomputed by WMMA GEMM) --------------------
__global__ void k_lstm1(const float* __restrict__ G1, const float* __restrict__ Whh,
                        float* __restrict__ hs1)
{
    __shared__ v2f   wt2[(H1/2) * 4*H1];   // transposed, float2-packed: wt2[j2*128+g]
    __shared__ v2f   h2[H1/2];
    __shared__ float gates[4*H1];
    int g = threadIdx.x;                    // 0..127
    #pragma unroll
    for (int j2 = 0; j2 < H1/2; j2++)
        wt2[j2*128 + g] = *(const v2f*)(Whh + g*H1 + 2*j2);
    if (g < H1) ((float*)h2)[g] = 0.f;
    float c = 0.f;
    __syncthreads();
    for (int t = 0; t < NN; t++) {
        float pre = G1[t*128 + g];
        #pragma unroll
        for (int j2 = 0; j2 < H1/2; j2++) {
            v2f w = wt2[j2*128 + g], hh = h2[j2];
            pre += w.x*hh.x + w.y*hh.y;
        }
        gates[g] = pre;
        __syncthreads();
        if (g < H1) {
            float i  = sigmoidf_(gates[g]);
            float f  = sigmoidf_(gates[H1 + g]);
            float gg = tanhf(gates[2*H1 + g]);
            float o  = sigmoidf_(gates[3*H1 + g]);
            c = f*c + i*gg;
            float hn = o * tanhf(c);
            ((float*)h2)[g] = hn;
            hs1[t*H1 + g] = hn;
        }
        __syncthreads();
    }
}

// ---------------- LSTM2 recurrence: Whh (256KB) resident in CDNA5's 320KB LDS ------------------
// x_t staging uses double-buffered async global->LDS prefetch when available.
__global__ void k_lstm2(const float* __restrict__ hs1, const float* __restrict__ Wih,
                        const float* __restrict__ Whh, const float* __restrict__ bih,
                        const float* __restrict__ bhh, float* __restrict__ hs2)
{
    extern __shared__ char smraw[];
    v2f*   wt2   = (v2f*)smraw;                            // [64][512] float2-packed transposed Whh
    v2f*   h2    = (v2f*)(smraw + (size_t)(H2/2)*4*H2*8);  // 64 float2
    float* gates = (float*)((char*)h2 + (H2/2)*8);         // 512
    float* xtbuf = gates + 4*H2;                           // 2 * 32 (double buffer)
    int g = threadIdx.x;                                   // 0..511
    for (int j2 = 0; j2 < H2/2; j2++)
        wt2[j2*512 + g] = *(const v2f*)(Whh + g*H2 + 2*j2);
    float wih[H1];
    #pragma unroll
    for (int j = 0; j < H1; j++) wih[j] = Wih[g*H1 + j];
    float bias = bih[g] + bhh[g];
    if (g < H2) ((float*)h2)[g] = 0.f;
    float c = 0.f;
#if ASYNC_XT
    if (g < H1)   // wave 0 only: prefetch x_0
        __builtin_amdgcn_global_load_async_to_lds_b32(
            (gptr_i)(hs1 + g), (lptr_i)(xtbuf + g), 0, 0);
#endif
    __syncthreads();
    for (int t = 0; t < NN; t++) {
        const int cur = (t & 1) * H1, nxt = ((t + 1) & 1) * H1;
#if ASYNC_XT
        if (g < H1) __builtin_amdgcn_s_wait_asynccnt(0);
        __syncthreads();
        if (g < H1 && t + 1 < NN)  // prefetch x_{t+1} into the other buffer
            __builtin_amdgcn_global_load_async_to_lds_b32(
                (gptr_i)(hs1 + (t+1)*H1 + g), (lptr_i)(xtbuf + nxt + g), 0, 0);
#else
        if (g < H1) xtbuf[cur + g] = hs1[t*H1 + g];
        __syncthreads();
#endif
        const float* xt = xtbuf + cur;
        float pre = bias;
        #pragma unroll
        for (int j = 0; j < H1; j++) pre += wih[j] * xt[j];
        #pragma unroll 8
        for (int j2 = 0; j2 < H2/2; j2++) {
            v2f w = wt2[j2*512 + g], hh = h2[j2];
            pre += w.x*hh.x + w.y*hh.y;
        }
        gates[g] = pre;
        __syncthreads();
        if (g < H2) {
            float i  = sigmoidf_(gates[g]);
            float f  = sigmoidf_(gates[H2 + g]);
            float gg = tanhf(gates[2*H2 + g]);
            float o  = sigmoidf_(gates[3*H2 + g]);
            c = f*c + i*gg;
            float hn = o * tanhf(c);
            ((float*)h2)[g] = hn;
            hs2[t*H2 + g] = hn;
        }
        __syncthreads();
    }
}

extern "C" void kernel_launch(void* const* d_in, const int* in_sizes, int n_in,
                              void* d_out, int out_size, void* d_ws, size_t ws_size,
                              hipStream_t stream)
{
    const float* x       = (const float*)d_in[0];
    const int*   ei      = (const int*)  d_in[1];
    const float* gat_W   = (const float*)d_in[2];
    const float* a_src   = (const float*)d_in[3];
    const float* a_dst   = (const float*)d_in[4];
    const float* gat_b   = (const float*)d_in[5];
    const float* l1_Wih  = (const float*)d_in[6];
    const float* l1_Whh  = (const float*)d_in[7];
    const float* l1_bih  = (const float*)d_in[8];
    const float* l1_bhh  = (const float*)d_in[9];
    const float* l2_Wih  = (const float*)d_in[10];
    const float* l2_Whh  = (const float*)d_in[11];
    const float* l2_bih  = (const float*)d_in[12];
    const float* l2_bhh  = (const float*)d_in[13];
    const float* lin_W   = (const float*)d_in[14];
    const float* lin_b   = (const float*)d_in[15];
    float* out = (float*)d_out;

    // workspace layout (~90 MB)
    float* ws = (float*)d_ws;
    size_t o = 0;
    float*    h     = ws + o; o += (size_t)NN*64;   // x@W
    float*    buf   = ws + o; o += (size_t)NN*64;   // GAT accum -> log_softmax output
    float*    as_   = ws + o; o += (size_t)NN*8;
    float*    ad_   = ws + o; o += (size_t)NN*8;
    unsigned* menc  = (unsigned*)(ws + o); o += (size_t)NN*8;
    float*    denom = ws + o; o += (size_t)NN*8;
    float*    G1    = ws + o; o += (size_t)NN*128;  // LSTM1 input proj + bias
    float*    hs1   = ws + o; o += (size_t)NN*32;
    float*    hs2   = ws + o; o += (size_t)NN*128;

    (void)hipMemsetAsync(buf,   0, (size_t)NN*64*sizeof(float), stream);
    (void)hipMemsetAsync(menc,  0, (size_t)NN*8*sizeof(unsigned), stream);
    (void)hipMemsetAsync(denom, 0, (size_t)NN*8*sizeof(float), stream);

    // GAT
    k_gemm_xw<<<( (NN/16)*4 + 7 )/8, 256, 0, stream>>>(x, gat_W, h);
    k_alpha<<<(NN*8 + 255)/256, 256, 0, stream>>>(h, a_src, a_dst, as_, ad_);
    k_edge_max<<<(NE + 255)/256, 256, 0, stream>>>(ei, as_, ad_, menc);
    k_edge_sum<<<(NE + 255)/256, 256, 0, stream>>>(ei, as_, ad_, menc, denom);
    k_edge_msg<<<(NE*8 + 255)/256, 256, 0, stream>>>(ei, as_, ad_, menc, denom, h, buf);
    k_finalize<<<(NN + 7)/8, 256, 0, stream>>>(buf, gat_b);

    // LSTM1: input projection as fully-unrolled WMMA GEMM (16 wmma/tile), then recurrence
    k_gemm_ABt<64><<<( (NN/16)*8 + 7 )/8, 256, 0, stream>>>(buf, l1_Wih, l1_bih, l1_bhh, G1, NN, 128);
    k_lstm1<<<1, 128, 0, stream>>>(G1, l1_Whh, hs1);

    // LSTM2: Whh in 320KB LDS (float2-packed), Wih row in registers, async x_t prefetch
    size_t lds2 = (size_t)(H2/2)*4*H2*8 + (H2/2)*8 + 4*H2*4 + 2*H1*4;  // 264960 B
    (void)hipFuncSetAttribute(reinterpret_cast<const void*>(k_lstm2),
                              hipFuncAttributeMaxDynamicSharedMemorySize, (int)lds2);
    k_lstm2<<<1, 512, lds2, stream>>>(hs1, l2_Wih, l2_Whh, l2_bih, l2_bhh, hs2);

    // final linear via fully-unrolled WMMA GEMM (32 wmma/tile)
    k_gemm_ABt<128><<<( (NN/16)*4 + 7 )/8, 256, 0, stream>>>(hs2, lin_W, lin_b, nullptr, out, NN, 64);
}